// SAGE_31662498906175
// MI455X (gfx1250) — compile-verified
//
#include <hip/hip_runtime.h>
#include <hip/hip_bf16.h>

// ---------------------------------------------------------------------------
// GraphSAGE forward for MI455X (gfx1250).
//   pre:    h0 = x @ W_pre + b_pre                      (fp32 WMMA GEMM)
//   layer:  agg = scatter_add(h[src] -> dst); mean = agg/max(cnt,1)
//           h' = relu(mean @ Wl + h @ Wr + bl)          (fused K=256 WMMA GEMM)
//   post:   log_softmax(h @ W_post + b_post)
// Bandwidth-bound on the edge scatter (~3.3 GB); GEMMs use native fp32 WMMA
// (V_WMMA_F32_16X16X4_F32) so precision matches the fp32 reference.
//
// W is staged in LDS in a WMMA-native quad-interleaved layout:
//   addr(k,n) = (k>>2)*512 + (n>>4)*64 + ((k>>1)&1)*32 + (n&15)*2 + (k&1)
// so each B operand (B[k][n], B[k+1][n]) is a single aligned ds_load_b64 and
// the two half-waves hit disjoint 32-bank halves (conflict-free, no padding).
// ---------------------------------------------------------------------------

typedef __attribute__((ext_vector_type(2))) float v2f;
typedef __attribute__((ext_vector_type(8))) float v8f;

#define H_DIM 128
#define STRIDE_A 36    // LDS row stride for A: {36*r % 64} distinct for r=0..15, 16B-aligned rows
#define KC 32          // K chunk streamed through LDS

// ---------------------------------------------------------------------------
// Fused GEMM:  C[M x 128] = relu?( A1*diag(1/max(cnt,1)) @ W1  +  A2 @ W2  + bias )
//   A1: M x K1 (row major), optional per-row scale from cnt (mean aggregation)
//   A2: M x K2 (row major), may be null (K2 == 0)
//   W1: K1 x 128, W2: K2 x 128 (row major), bias: 128
// Block: 256 threads (8 waves), tile M=128 rows x N=128 cols.
// Wave w computes rows [16w, 16w+16) x all 128 cols: 8 accum tiles of 16x16.
// ---------------------------------------------------------------------------
__global__ void sage_gemm128(const float* __restrict__ A1,
                             const float* __restrict__ cnt,   // null => no scaling
                             const float* __restrict__ A2,    // null => K2 == 0
                             const float* __restrict__ W1,
                             const float* __restrict__ W2,
                             const float* __restrict__ bias,
                             float* __restrict__ C,
                             int M, int K1, int K2, int relu)
{
    extern __shared__ float smem[];
    const int Ktot = K1 + K2;
    float* lds_w = smem;                       // Ktot * 128 floats, quad-interleaved
    float* lds_a = smem + Ktot * H_DIM;        // 128  * STRIDE_A floats

    const int tid      = threadIdx.x;
    const int w        = tid >> 5;
    const int lane     = tid & 31;
    const int l16      = lane & 15;
    const int lh       = lane >> 4;            // half-wave select (K offset +0 / +2)
    const int blockRow = blockIdx.x * 128;

    // ---- load fused weight [Ktot x 128] into LDS (quad-interleaved) ----
    for (int idx = tid; idx < Ktot * H_DIM; idx += 256) {
        const int k = idx >> 7;
        const int n = idx & 127;
        const float v = (k < K1) ? W1[idx] : W2[idx - K1 * H_DIM];
        lds_w[((k >> 2) << 9) + ((n >> 4) << 6) + (((k >> 1) & 1) << 5)
              + ((n & 15) << 1) + (k & 1)] = v;
    }

    v8f acc[8] = {};

    // ---- stream A in KC-wide chunks, accumulate with fp32 WMMA ----
    for (int kc = 0; kc < Ktot; kc += KC) {
        // cooperative A-chunk load: 128 rows x 32 cols, float4 per thread-iter
        {
            const bool fromA1 = (kc < K1);
            const float* Ap = fromA1 ? A1 : A2;
            const int    K  = fromA1 ? K1 : K2;
            const int    kb = fromA1 ? kc : (kc - K1);
            const int r0 = tid >> 3;           // 0..31
            const int cg = (tid & 7) * 4;      // 0,4,...,28
            for (int ri = r0; ri < 128; ri += 32) {
                const int row = blockRow + ri;
                float4 v = make_float4(0.f, 0.f, 0.f, 0.f);
                if (row < M) {
                    v = *(const float4*)(Ap + (size_t)row * K + kb + cg);
                    if (fromA1 && cnt) {
                        const float s = 1.0f / fmaxf(cnt[row], 1.0f);
                        v.x *= s; v.y *= s; v.z *= s; v.w *= s;
                    }
                }
                *(float4*)&lds_a[ri * STRIDE_A + cg] = v;
            }
        }
        __syncthreads();

        const float* aw    = &lds_a[(w * 16 + l16) * STRIDE_A + lh * 2];
        const float* wbase = &lds_w[lh * 32 + l16 * 2];
#pragma unroll
        for (int k4 = 0; k4 < KC / 4; ++k4) {
            // A operand (16x4 f32): lanes 0-15 K={k,k+1}, lanes 16-31 K={k+2,k+3}
            v2f a = *(const v2f*)(aw + k4 * 4);
            const float* wq = wbase + (((kc + k4 * 4) >> 2) << 9);
            v2f b[8];
#pragma unroll
            for (int nt = 0; nt < 8; ++nt)
                b[nt] = *(const v2f*)(wq + nt * 64);   // single b64, conflict-free
#pragma unroll
            for (int nt = 0; nt < 8; ++nt)
                acc[nt] = __builtin_amdgcn_wmma_f32_16x16x4_f32(
                    false, a, false, b[nt], (short)0, acc[nt], false, false);
        }
        __syncthreads();
    }

    // ---- epilogue: bias, relu, store ----
#pragma unroll
    for (int nt = 0; nt < 8; ++nt) {
        const int col = nt * 16 + l16;
        const float bv = bias ? bias[col] : 0.0f;
#pragma unroll
        for (int r = 0; r < 8; ++r) {
            const int row = blockRow + w * 16 + r + lh * 8;   // C/D VGPR layout
            if (row < M) {
                float v = acc[nt][r] + bv;
                if (relu) v = fmaxf(v, 0.0f);
                C[(size_t)row * H_DIM + col] = v;
            }
        }
    }
}

// one thread per edge: cnt[dst] += 1
__global__ void sage_degree(const int* __restrict__ dst, float* __restrict__ cnt, int E)
{
    const int e = blockIdx.x * blockDim.x + threadIdx.x;
    if (e < E) atomicAdd(&cnt[dst[e]], 1.0f);
}

// one wave per edge (32 lanes x float4 = one 128-float row), coalesced gather
// + global_atomic_add_f32 scatter
__global__ void sage_scatter(const float* __restrict__ h,
                             const int* __restrict__ src,
                             const int* __restrict__ dst,
                             float* __restrict__ agg, int E)
{
    const long long tid   = (long long)blockIdx.x * blockDim.x + threadIdx.x;
    const long long total = (long long)E * 32;
    if (tid >= total) return;
    const int e  = (int)(tid >> 5);
    const int ch = ((int)tid & 31) * 4;
    const int s = src[e];
    const int d = dst[e];
    const float4 v = *(const float4*)(h + (size_t)s * H_DIM + ch);
    float* ap = agg + (size_t)d * H_DIM + ch;
    atomicAdd(ap + 0, v.x);
    atomicAdd(ap + 1, v.y);
    atomicAdd(ap + 2, v.z);
    atomicAdd(ap + 3, v.w);
}

// post linear (128 -> 40) + log_softmax; one wave32 per node, W_post in LDS.
__global__ void sage_post_lsm(const float* __restrict__ h,
                              const float* __restrict__ Wp,
                              const float* __restrict__ bp,
                              float* __restrict__ out, int M)
{
    __shared__ float lw[H_DIM * 40];
    __shared__ float lb[40];
    for (int i = threadIdx.x; i < H_DIM * 40; i += blockDim.x) lw[i] = Wp[i];
    for (int i = threadIdx.x; i < 40; i += blockDim.x) lb[i] = bp[i];
    __syncthreads();

    const int wave = threadIdx.x >> 5;
    const int lane = threadIdx.x & 31;
    const int node = blockIdx.x * 8 + wave;
    if (node >= M) return;

    const float* hp = h + (size_t)node * H_DIM;
    const int c0 = lane;            // < 40 always
    const int c1 = lane + 32;       // valid iff lane < 8
    float a0 = 0.0f, a1 = 0.0f;
#pragma unroll 4
    for (int k = 0; k < H_DIM; ++k) {
        const float hk = hp[k];
        a0 = fmaf(hk, lw[k * 40 + c0], a0);
        if (lane < 8) a1 = fmaf(hk, lw[k * 40 + c1], a1);
    }
    const float NEG = -1e30f;
    float v0 = a0 + lb[c0];
    float v1 = (lane < 8) ? (a1 + lb[c1]) : NEG;

    float m = fmaxf(v0, v1);
#pragma unroll
    for (int off = 16; off > 0; off >>= 1) m = fmaxf(m, __shfl_xor(m, off, 32));
    float e = __expf(v0 - m) + ((lane < 8) ? __expf(v1 - m) : 0.0f);
#pragma unroll
    for (int off = 16; off > 0; off >>= 1) e += __shfl_xor(e, off, 32);
    const float ls = __logf(e);

    out[(size_t)node * 40 + c0] = v0 - m - ls;
    if (lane < 8) out[(size_t)node * 40 + c1] = v1 - m - ls;
}

extern "C" void kernel_launch(void* const* d_in, const int* in_sizes, int n_in,
                              void* d_out, int out_size, void* d_ws, size_t ws_size,
                              hipStream_t stream)
{
    (void)n_in; (void)out_size; (void)ws_size;
    const float* x      = (const float*)d_in[0];
    const int*   ei     = (const int*)  d_in[1];
    const float* W_pre  = (const float*)d_in[2];
    const float* b_pre  = (const float*)d_in[3];
    const float* Wl0    = (const float*)d_in[4];
    const float* bl0    = (const float*)d_in[5];
    const float* Wr0    = (const float*)d_in[6];
    const float* Wl1    = (const float*)d_in[7];
    const float* bl1    = (const float*)d_in[8];
    const float* Wr1    = (const float*)d_in[9];
    const float* W_post = (const float*)d_in[10];
    const float* b_post = (const float*)d_in[11];

    const int N = in_sizes[0] / 256;      // 100000 nodes, IN=256
    const int E = in_sizes[1] / 2;        // 1.6M edges
    const int* src = ei;
    const int* dst = ei + E;

    float* ws  = (float*)d_ws;
    float* h0  = ws;                              // N*128
    float* h1  = h0 + (size_t)N * H_DIM;          // N*128
    float* agg = h1 + (size_t)N * H_DIM;          // N*128
    float* cnt = agg + (size_t)N * H_DIM;         // N

    const dim3 blk(256);
    const int gemmBlocks = (N + 127) / 128;
    const size_t ldsBytes = (size_t)(256 * H_DIM + 128 * STRIDE_A) * sizeof(float);
    const int degBlocks  = (E + 255) / 256;
    const int scatBlocks = (int)(((long long)E * 32 + 255) / 256);
    const int postBlocks = (N + 7) / 8;

    // pre: h0 = x @ W_pre + b_pre   (K = 256)
    sage_gemm128<<<gemmBlocks, blk, ldsBytes, stream>>>(
        x, nullptr, nullptr, W_pre, nullptr, b_pre, h0, N, 256, 0, 0);

    // degrees (graph is fixed across layers)
    hipMemsetAsync(cnt, 0, (size_t)N * sizeof(float), stream);
    sage_degree<<<degBlocks, blk, 0, stream>>>(dst, cnt, E);

    // layer 0
    hipMemsetAsync(agg, 0, (size_t)N * H_DIM * sizeof(float), stream);
    sage_scatter<<<scatBlocks, blk, 0, stream>>>(h0, src, dst, agg, E);
    sage_gemm128<<<gemmBlocks, blk, ldsBytes, stream>>>(
        agg, cnt, h0, Wl0, Wr0, bl0, h1, N, H_DIM, H_DIM, 1);

    // layer 1
    hipMemsetAsync(agg, 0, (size_t)N * H_DIM * sizeof(float), stream);
    sage_scatter<<<scatBlocks, blk, 0, stream>>>(h1, src, dst, agg, E);
    sage_gemm128<<<gemmBlocks, blk, ldsBytes, stream>>>(
        agg, cnt, h1, Wl1, Wr1, bl1, h0, N, H_DIM, H_DIM, 1);

    // post linear + log_softmax
    sage_post_lsm<<<postBlocks, blk, 0, stream>>>(h0, W_post, b_post, (float*)d_out, N);
}